// ConvSSMBlock_13091060318317
// MI455X (gfx1250) — compile-verified
//
#include <hip/hip_runtime.h>
#include <math.h>

// ConvSSM forward on MI455X (gfx1250).
// h_c = IDFT2( M_c * DFT2(x_c) ),  M_c = (sum_{j=0}^{7} Afft_c^j) * Bfft_c.
// Every 128-point DFT is a dense 128x128 FP32 matmul on the WMMA f32 path
// (v_wmma_f32_16x16x4_f32). Stages are fused into 4 GEMM kernels:
//   fwd cols (dual-A, shared x panel) -> complex rows -> pointwise M ->
//   complex inverse cols -> real inverse rows (NHWC scatter).
// B panels are staged into LDS with gfx1250 async-to-LDS instructions
// (ASYNCcnt), inner loop reads them with ds loads.

typedef __attribute__((ext_vector_type(2))) float v2f;
typedef __attribute__((ext_vector_type(8))) float v8f;

#define BATCH 4
#define HH 128
#define WW 128
#define CC 192
#define NPLANES (BATCH * CC)            // 768
#define PLANE_ELEMS (HH * WW)           // 16384
#define XPS ((long long)HH * WW * CC)   // NHWC batch stride (3145728)
#define XRS ((long long)WW * CC)        // NHWC row stride (24576)

// Flip to 0 if the assembler rejects the async-to-LDS mnemonics.
#define USE_ASYNC_LDS 1

// Workspace layout (floats):
static const long long OFF_FR = 0;                         // 16384
static const long long OFF_FI = 16384;                     // 16384
static const long long OFF_M  = 32768;                     // 6291456 (float2)
static const long long OFF_YR = 6324224;                   // 768*16384
static const long long OFF_YI = 18907136;
static const long long OFF_ZR = 31490048;
static const long long OFF_ZI = 44072960;                  // end ~227 MB

// ---------------------------------------------------------------------------
// Helpers
__device__ __forceinline__ v2f load2(const float* p) {
  float2 t = *(const float2*)p;
  v2f r;
  r.x = t.x;
  r.y = t.y;
  return r;
}

// LDS B-fragment: B[k][col] with compile-time strides.
template <int LRS, int LCS>
__device__ __forceinline__ v2f ldsB(const float* LB, int k, int c) {
  v2f r;
  if (LRS == 1) {  // k-pairs contiguous -> single 8B LDS load
    float2 t = *(const float2*)&LB[c * LCS + k];
    r.x = t.x;
    r.y = t.y;
  } else {
    r.x = LB[k * LRS + c * LCS];
    r.y = LB[(k + 1) * LRS + c * LCS];
  }
  return r;
}

__device__ __forceinline__ v8f wmma4(v2f a, v2f b, v8f c) {
  return __builtin_amdgcn_wmma_f32_16x16x4_f32(false, a, false, b, (short)0, c,
                                               false, false);
}

// Contiguous 64KB panel -> LDS (async DMA path, bypasses VGPRs).
__device__ __forceinline__ void stage_contig(float* LB, const float* src,
                                             int tid) {
#if USE_ASYNC_LDS
  unsigned lbase = (unsigned)(size_t)LB;  // flat addr[31:0] == LDS offset
  for (int e = tid; e < PLANE_ELEMS / 4; e += 512) {
    asm volatile("global_load_async_to_lds_b128 %0, %1, off" ::"v"(
                     lbase + (unsigned)e * 16u),
                 "v"(src + e * 4)
                 : "memory");
  }
#else
  float4* l = (float4*)LB;
  const float4* g = (const float4*)src;
  for (int e = tid; e < PLANE_ELEMS / 4; e += 512) l[e] = g[e];
#endif
}

// NHWC gather: LB[k*128+col] = src[k*XRS + col*CC]
__device__ __forceinline__ void stage_gather(float* LB, const float* src,
                                             int tid) {
#if USE_ASYNC_LDS
  unsigned lbase = (unsigned)(size_t)LB;
  for (int e = tid; e < PLANE_ELEMS; e += 512) {
    int k = e >> 7, col = e & 127;
    asm volatile("global_load_async_to_lds_b32 %0, %1, off" ::"v"(
                     lbase + (unsigned)e * 4u),
                 "v"(src + (long long)k * XRS + (long long)col * CC)
                 : "memory");
  }
#else
  for (int e = tid; e < PLANE_ELEMS; e += 512) {
    int k = e >> 7, col = e & 127;
    LB[e] = src[(long long)k * XRS + (long long)col * CC];
  }
#endif
}

__device__ __forceinline__ void stage_wait() {
#if USE_ASYNC_LDS
  asm volatile("s_wait_asynccnt 0x0" ::: "memory");
#endif
  __syncthreads();
}

// ---------------------------------------------------------------------------
// DFT matrix: F[u][t] = exp(-2*pi*i*u*t/128); angle reduced mod 128 exactly.
__global__ __launch_bounds__(256) void init_F_kernel(float* __restrict__ Fr,
                                                     float* __restrict__ Fi) {
  int idx = blockIdx.x * 256 + threadIdx.x;  // < 16384
  int u = idx >> 7, t = idx & 127;
  int k = (u * t) & 127;
  float ang = -6.28318530717958647692f * (float)k / 128.0f;
  float s, c;
  sincosf(ang, &s, &c);
  Fr[idx] = c;
  Fi[idx] = s;
}

// ---------------------------------------------------------------------------
// M_c[u,v] = (sum_{j=0}^{7} Afft^j) * Bfft, via LDS twiddle table.
__global__ __launch_bounds__(256) void build_M_kernel(
    const float* __restrict__ Ak, const float* __restrict__ Bk,
    float2* __restrict__ M) {
  __shared__ float twc[128];
  __shared__ float tws[128];
  if (threadIdx.x < 128) {
    float ang = -6.28318530717958647692f * (float)threadIdx.x / 128.0f;
    float s, c;
    sincosf(ang, &s, &c);
    twc[threadIdx.x] = c;
    tws[threadIdx.x] = s;
  }
  __syncthreads();

  int idx = blockIdx.x * 256 + threadIdx.x;  // < 192*16384
  int cch = idx >> 14;
  int uv = idx & 16383;
  int u = uv >> 7;
  int v = uv & 127;

  float dec[7];
#pragma unroll
  for (int p = 0; p < 7; ++p) dec[p] = __expf(-0.3f * (float)p);

  const float* Ac = Ak + cch * 49;
  const float* Bc = Bk + cch * 49;
  float ar = 0.f, ai = 0.f, br = 0.f, bi = 0.f;
#pragma unroll
  for (int p = 0; p < 7; ++p) {
#pragma unroll
    for (int q = 0; q < 7; ++q) {
      int k = (u * p + v * q) & 127;
      float cw = twc[k], sw = tws[k];
      float a = Ac[p * 7 + q] * dec[p] * dec[q];
      float b = Bc[p * 7 + q];
      ar += a * cw;
      ai += a * sw;
      br += b * cw;
      bi += b * sw;
    }
  }
  float sr = 1.f, si = 0.f, pr = 1.f, pi = 0.f;
#pragma unroll
  for (int j = 1; j < 8; ++j) {
    float nr = pr * ar - pi * ai;
    float ni = pr * ai + pi * ar;
    pr = nr;
    pi = ni;
    sr += pr;
    si += pi;
  }
  M[idx] = make_float2(sr * br - si * bi, sr * bi + si * br);
}

// ---------------------------------------------------------------------------
// Frequency-domain pointwise multiply: (Zr,Zi) *= M_c, in place.
__global__ __launch_bounds__(256) void pointwise_M_kernel(
    float* __restrict__ Zr, float* __restrict__ Zi,
    const float2* __restrict__ M) {
  int idx = blockIdx.x * 256 + threadIdx.x;  // < 768*16384
  int plane = idx >> 14;
  int cch = plane % CC;
  int e = idx & 16383;
  float zr = Zr[idx], zi = Zi[idx];
  float2 m = M[cch * PLANE_ELEMS + e];
  Zr[idx] = zr * m.x - zi * m.y;
  Zi[idx] = zr * m.y + zi * m.x;
}

// ---------------------------------------------------------------------------
// Fragment layouts per CDNA5 ISA 7.12.2 (wave32):
//   A 16x4 f32: lanes 0-15 -> {K,K+1}, lanes 16-31 -> {K+2,K+3}, M = lane&15
//   B 4x16 f32: mirrored (lane half selects K pair), N = lane&15
//   C/D 16x16:  VGPR r -> M = r + 8*(lane>=16), N = lane&15

// Forward columns, fused: Yr = Fr*Xp, Yi = Fi*Xp (one shared x panel in LDS).
// One block (512 thr = 16 waves) per plane; wave w -> tile row w>>1,
// 4 N-tiles starting at 4*(w&1).
__global__ __launch_bounds__(512) void fwd_cols_kernel(
    const float* __restrict__ Fr, const float* __restrict__ Fi,
    const float* __restrict__ x, float* __restrict__ Yr,
    float* __restrict__ Yi) {
  __shared__ __align__(16) float LB[PLANE_ELEMS];
  const int plane = blockIdx.x;
  const float* xp = x + (long long)(plane / CC) * XPS + (plane % CC);
  stage_gather(LB, xp, threadIdx.x);
  stage_wait();

  const int wv = threadIdx.x >> 5, lane = threadIdx.x & 31;
  const int l15 = lane & 15, hi = lane >> 4;
  const int tr = wv >> 1, tc0 = (wv & 1) * 4;
  const int arow = tr * 16 + l15;

  v8f ar[4] = {}, ai[4] = {};
#pragma unroll 2
  for (int k0 = 0; k0 < 128; k0 += 4) {
    const int kk = k0 + (hi << 1);
    v2f afr = load2(&Fr[arow * 128 + kk]);
    v2f afi = load2(&Fi[arow * 128 + kk]);
#pragma unroll
    for (int j = 0; j < 4; ++j) {
      const int col = (tc0 + j) * 16 + l15;
      v2f bv = ldsB<128, 1>(LB, kk, col);
      ar[j] = wmma4(afr, bv, ar[j]);
      ai[j] = wmma4(afi, bv, ai[j]);
    }
  }
  float* dr = Yr + (long long)plane * PLANE_ELEMS;
  float* di = Yi + (long long)plane * PLANE_ELEMS;
#pragma unroll
  for (int j = 0; j < 4; ++j) {
    const int col = (tc0 + j) * 16 + l15;
#pragma unroll
    for (int r = 0; r < 8; ++r) {
      const int row = tr * 16 + r + (hi << 3);
      dr[row * 128 + col] = ar[j][r];
      di[row * 128 + col] = ai[j][r];
    }
  }
}

// Complex GEMM, fused: Dr = Ar*Br - sigma*Ai*Bi ; Di = Ar*Bi + sigma*Ai*Br.
// sigma=+1: multiply by F^T (forward rows); sigma=-1: multiply by conj(F)
// (inverse columns). Two blocks per plane (half = 4 tile rows); wave w ->
// tile row half*4 + (w>>2), 2 N-tiles starting at 2*(w&3).
// LRS/LCS: LDS strides of staged B panels (raw F panels read transposed:
// LRS=1/LCS=128; row-major Z panels: LRS=128/LCS=1).
template <int LRS, int LCS>
__global__ __launch_bounds__(512) void cgemm_kernel(
    const float* __restrict__ Arp, const float* __restrict__ Aip,
    long long aPos, const float* __restrict__ Brp,
    const float* __restrict__ Bip, long long bPos, float* __restrict__ Drp,
    float* __restrict__ Dip, float sigma) {
  __shared__ __align__(16) float LBr[PLANE_ELEMS];
  __shared__ __align__(16) float LBi[PLANE_ELEMS];
  const int plane = blockIdx.x >> 1;
  const int half = blockIdx.x & 1;
  stage_contig(LBr, Brp + (long long)plane * bPos, threadIdx.x);
  stage_contig(LBi, Bip + (long long)plane * bPos, threadIdx.x);
  stage_wait();

  const int wv = threadIdx.x >> 5, lane = threadIdx.x & 31;
  const int l15 = lane & 15, hi = lane >> 4;
  const int tr = half * 4 + (wv >> 2);
  const int tc0 = (wv & 3) * 2;
  const int arow = tr * 16 + l15;

  const float* Ar = Arp + (long long)plane * aPos;
  const float* Ai = Aip + (long long)plane * aPos;

  v8f rr[2] = {}, ii[2] = {}, ri[2] = {}, ir[2] = {};
#pragma unroll 2
  for (int k0 = 0; k0 < 128; k0 += 4) {
    const int kk = k0 + (hi << 1);
    v2f avr = load2(&Ar[arow * 128 + kk]);
    v2f avi = load2(&Ai[arow * 128 + kk]);
#pragma unroll
    for (int j = 0; j < 2; ++j) {
      const int col = (tc0 + j) * 16 + l15;
      v2f bvr = ldsB<LRS, LCS>(LBr, kk, col);
      v2f bvi = ldsB<LRS, LCS>(LBi, kk, col);
      rr[j] = wmma4(avr, bvr, rr[j]);
      ii[j] = wmma4(avi, bvi, ii[j]);
      ri[j] = wmma4(avr, bvi, ri[j]);
      ir[j] = wmma4(avi, bvr, ir[j]);
    }
  }
  float* dr = Drp + (long long)plane * PLANE_ELEMS;
  float* di = Dip + (long long)plane * PLANE_ELEMS;
#pragma unroll
  for (int j = 0; j < 2; ++j) {
    const int col = (tc0 + j) * 16 + l15;
#pragma unroll
    for (int r = 0; r < 8; ++r) {
      const int row = tr * 16 + r + (hi << 3);
      dr[row * 128 + col] = rr[j][r] - sigma * ii[j][r];
      di[row * 128 + col] = ri[j][r] + sigma * ir[j][r];
    }
  }
}

// Inverse rows (real output): h = (Ur*FrT + Ui*FiT) / 16384 -> NHWC scatter.
__global__ __launch_bounds__(512) void inv_rows_kernel(
    const float* __restrict__ Urp, const float* __restrict__ Uip,
    const float* __restrict__ Fr, const float* __restrict__ Fi,
    float* __restrict__ out) {
  __shared__ __align__(16) float LBr[PLANE_ELEMS];
  __shared__ __align__(16) float LBi[PLANE_ELEMS];
  const int plane = blockIdx.x;
  stage_contig(LBr, Fr, threadIdx.x);
  stage_contig(LBi, Fi, threadIdx.x);
  stage_wait();

  const int wv = threadIdx.x >> 5, lane = threadIdx.x & 31;
  const int l15 = lane & 15, hi = lane >> 4;
  const int tr = wv >> 1, tc0 = (wv & 1) * 4;
  const int arow = tr * 16 + l15;

  const float* Ur = Urp + (long long)plane * PLANE_ELEMS;
  const float* Ui = Uip + (long long)plane * PLANE_ELEMS;

  v8f srr[4] = {}, sii[4] = {};
#pragma unroll 2
  for (int k0 = 0; k0 < 128; k0 += 4) {
    const int kk = k0 + (hi << 1);
    v2f avr = load2(&Ur[arow * 128 + kk]);
    v2f avi = load2(&Ui[arow * 128 + kk]);
#pragma unroll
    for (int j = 0; j < 4; ++j) {
      const int col = (tc0 + j) * 16 + l15;
      v2f bvr = ldsB<1, 128>(LBr, kk, col);  // FrT
      v2f bvi = ldsB<1, 128>(LBi, kk, col);  // FiT
      srr[j] = wmma4(avr, bvr, srr[j]);
      sii[j] = wmma4(avi, bvi, sii[j]);
    }
  }
  float* d = out + (long long)(plane / CC) * XPS + (plane % CC);
  const float scale = 1.0f / 16384.0f;
#pragma unroll
  for (int j = 0; j < 4; ++j) {
    const int col = (tc0 + j) * 16 + l15;
#pragma unroll
    for (int r = 0; r < 8; ++r) {
      const int row = tr * 16 + r + (hi << 3);
      d[(long long)row * XRS + (long long)col * CC] =
          scale * (srr[j][r] + sii[j][r]);
    }
  }
}

// ---------------------------------------------------------------------------
extern "C" void kernel_launch(void* const* d_in, const int* in_sizes, int n_in,
                              void* d_out, int out_size, void* d_ws,
                              size_t ws_size, hipStream_t stream) {
  (void)in_sizes; (void)n_in; (void)out_size; (void)ws_size;
  const float* x  = (const float*)d_in[0];   // (4,128,128,192) NHWC
  const float* Ak = (const float*)d_in[1];   // (192,7,7)
  const float* Bk = (const float*)d_in[2];   // (192,7,7)
  float* out = (float*)d_out;                // (4,128,128,192) NHWC
  float* w = (float*)d_ws;

  float* Fr = w + OFF_FR;
  float* Fi = w + OFF_FI;
  float2* M = (float2*)(w + OFF_M);
  float* Yr = w + OFF_YR;
  float* Yi = w + OFF_YI;
  float* Zr = w + OFF_ZR;
  float* Zi = w + OFF_ZI;

  // Precompute DFT matrix and frequency response M.
  init_F_kernel<<<PLANE_ELEMS / 256, 256, 0, stream>>>(Fr, Fi);
  build_M_kernel<<<CC * PLANE_ELEMS / 256, 256, 0, stream>>>(Ak, Bk, M);

  // Forward columns: (Yr, Yi) = (Fr, Fi) * Xp   (single x gather per plane)
  fwd_cols_kernel<<<NPLANES, 512, 0, stream>>>(Fr, Fi, x, Yr, Yi);
  // Forward rows: Z = Y * F^T  (complex, sigma=+1; F panels read transposed)
  cgemm_kernel<1, 128><<<NPLANES * 2, 512, 0, stream>>>(
      Yr, Yi, PLANE_ELEMS, Fr, Fi, 0, Zr, Zi, +1.f);
  // Frequency-domain multiply by M (in place on Z).
  pointwise_M_kernel<<<NPLANES * PLANE_ELEMS / 256, 256, 0, stream>>>(Zr, Zi,
                                                                      M);
  // Inverse columns: U = conj(F) * W  (complex, sigma=-1) -> Y buffers
  cgemm_kernel<128, 1><<<NPLANES * 2, 512, 0, stream>>>(
      Fr, Fi, 0, Zr, Zi, PLANE_ELEMS, Yr, Yi, -1.f);
  // Inverse rows: h = (Ur*FrT + Ui*FiT) / 16384 -> NHWC output
  inv_rows_kernel<<<NPLANES, 512, 0, stream>>>(Yr, Yi, Fr, Fi, out);
}